// SEGNNUpdate_30915174596962
// MI455X (gfx1250) — compile-verified
//
#include <hip/hip_runtime.h>
#include <math.h>

#define NS    64
#define NVEC  32
#define DIMF  160          // NS + 3*NV
#define KTOT  192          // stacked K: 96 (tp) + 64 + 32
#define KPAD  196          // padded LDS row stride (dwords) to avoid bank conflicts
#define NCOL  128          // 96 h_s columns + 32 h_v columns
#define WAVES 8
#define TILE_M 16
#define NODES_PER_BLOCK (WAVES * TILE_M)

typedef __attribute__((ext_vector_type(2))) float v2f;
typedef __attribute__((ext_vector_type(8))) float v8f;

#define WMMA_F32_16x16x4(a, b, c) \
  __builtin_amdgcn_wmma_f32_16x16x4_f32(false, (a), false, (b), (short)0, (c), false, false)

#if defined(__has_builtin)
#if __has_builtin(__builtin_amdgcn_global_load_async_to_lds_b32)
#define HAVE_ASYNC_LDS 1
#endif
#endif

typedef __attribute__((address_space(1))) int gint_as1;   // global (AS1) int
typedef __attribute__((address_space(3))) int lint_as3;   // LDS (AS3) int

__device__ __forceinline__ float sigmoid_(float x) { return 1.0f / (1.0f + __expf(-x)); }

// ---------------- kernel 1: zero the message accumulator ----------------
__global__ void zero_ws_kernel(float* __restrict__ p, int n) {
  int i = blockIdx.x * blockDim.x + threadIdx.x;
  if (i < n) p[i] = 0.0f;
}

// ---------------- kernel 2: segment-sum (scatter-add) -------------------
__global__ void edge_scatter_kernel(const float* __restrict__ em,
                                    const int* __restrict__ dst,
                                    float* __restrict__ msg, int E) {
  int idx = blockIdx.x * blockDim.x + threadIdx.x;
  int total = E * DIMF;
  if (idx >= total) return;
  int e = idx / DIMF;
  int f = idx - e * DIMF;
  const float s5 = 0.4472135954999579f;  // 1/sqrt(AVG_NEI)
  atomicAdd(&msg[dst[e] * DIMF + f], em[idx] * s5);
}

// ---------------- kernel 3: fused node update via fp32 WMMA -------------
__global__ void node_update_kernel(const float* __restrict__ feats,
                                   const float* __restrict__ attrs,
                                   const float* __restrict__ msg,
                                   const float* __restrict__ w00,
                                   const float* __restrict__ w01,
                                   const float* __restrict__ w10,
                                   const float* __restrict__ w11,
                                   const float* __restrict__ Wls,
                                   const float* __restrict__ Wlv,
                                   const float* __restrict__ W000,
                                   const float* __restrict__ W110,
                                   const float* __restrict__ W011,
                                   const float* __restrict__ W101,
                                   float* __restrict__ out, int N) {
  extern __shared__ float smem[];
  float* Wt  = smem;                    // [NCOL][KPAD]  transposed stacked weights
  float* Ast = smem + NCOL * KPAD;      // [WAVES][TILE_M][KPAD] per-wave A staging

  const int tid = threadIdx.x;

  // ---- stage stacked, transposed weight matrix Wt[col][k] ----
  // Async LDS copies (ASYNCcnt-tracked): issue the transposing gather now and
  // overlap it with the A_s build below, which does not touch Wt.
  for (int idx = tid; idx < NCOL * KTOT; idx += blockDim.x) {
    int col = idx / KTOT;
    int k   = idx - col * KTOT;
    const float* src;
    if (col < 96) {                      // h_s columns
      if (k < 96)       src = &Wls[k * 96 + col];
      else if (k < 160) src = &W000[(k - 96) * 96 + col];
      else              src = &W110[(k - 160) * 96 + col];
    } else {                             // h_v columns
      int w = col - 96;
      if (k < 96)       src = &Wlv[k * 32 + w];
      else if (k < 160) src = &W011[(k - 96) * 32 + w];
      else              src = &W101[(k - 160) * 32 + w];
    }
#if defined(HAVE_ASYNC_LDS)
    __builtin_amdgcn_global_load_async_to_lds_b32(
        (gint_as1*)(int*)(float*)src,
        (lint_as3*)(int*)&Wt[col * KPAD + k],
        0, 0);
#else
    Wt[col * KPAD + k] = *src;
#endif
  }

  const int wave = tid >> 5;
  const int lane = tid & 31;
  const int r    = lane & 15;            // row within 16-node tile
  const int half = lane >> 4;            // 0: K lo / cols 0..95, 1: K hi / cols 96..191
  float* A = Ast + wave * TILE_M * KPAD;

  const int tileBase = blockIdx.x * NODES_PER_BLOCK + wave * TILE_M;
  const int node     = tileBase + r;
  const bool valid   = node < N;

  // scale constants (folded into A operand)
  const float INV_SQRT3 = 0.5773502691896258f;
  const float inv_fan   = 0.10206207261596575f;               // 1/sqrt(96)
  const float inv2      = 0.7071067811865476f;                // 1/sqrt(2)
  const float invs32    = 0.17677669529663687f;               // 1/sqrt(32)
  const float c000      = inv2 * 0.125f;                      // inv2/sqrt(64)
  const float c110      = INV_SQRT3 * inv2 * invs32;
  const float c101      = inv2 * invs32;

  float a_s = 0.f, av0 = 0.f, av1 = 0.f, av2 = 0.f;
  const float* mrow = msg   + (long)node * DIMF;   // only deref'd when valid
  const float* frow = feats + (long)node * DIMF;
  if (valid) {
    a_s = attrs[node * 4 + 0];
    av0 = attrs[node * 4 + 1];
    av1 = attrs[node * 4 + 2];
    av2 = attrs[node * 4 + 3];
    // pull this node's rows toward the near caches (re-read by 4 build phases)
    __builtin_prefetch(mrow, 0, 3);
    __builtin_prefetch(mrow + 112, 0, 3);
    __builtin_prefetch(frow, 0, 3);
    __builtin_prefetch(frow + 112, 0, 3);
  }

  // ---- build A_s rows: [tp_s*inv_fan | xs*a_s*c000 | (xv.a_v)*c110] ----
  for (int j = 0; j < 96; ++j) {
    int col = half * 96 + j;
    float v = 0.f;
    if (valid) {
      if (col < 64) {
        v = mrow[col] * a_s * w00[col] * inv_fan;
      } else if (col < 96) {
        int u = col - 64;
        float d = mrow[64 + u * 3] * av0 + mrow[64 + u * 3 + 1] * av1 + mrow[64 + u * 3 + 2] * av2;
        v = d * (INV_SQRT3 * inv_fan) * w11[u];
      } else if (col < 160) {
        int u = col - 96;
        v = frow[u] * a_s * c000;
      } else {
        int u = col - 160;
        float d = frow[64 + u * 3] * av0 + frow[64 + u * 3 + 1] * av1 + frow[64 + u * 3 + 2] * av2;
        v = d * c110;
      }
    }
    A[r * KPAD + col] = v;
  }

#if defined(HAVE_ASYNC_LDS)
  // weight gather must have landed in LDS before anyone reads Wt
#if defined(__has_builtin) && __has_builtin(__builtin_amdgcn_s_wait_asynccnt)
  __builtin_amdgcn_s_wait_asynccnt(0);
#else
  asm volatile("s_wait_asynccnt 0x0" ::: "memory");
#endif
#endif
  __syncthreads();

  // ---- h_s = A_s @ Wt[:,0..95]  (6 column tiles, 48 K-steps of fp32 WMMA) ----
  const int afo = r * KPAD + half * 2;      // A fragment: lanes 0-15 {k,k+1}, 16-31 {k+2,k+3}
  const int bl  = lane & 15;                // B fragment column within tile

  v8f hs[6] = {};
  for (int k0 = 0; k0 < KTOT; k0 += 4) {
    v2f a = *(const v2f*)&A[afo + k0];
    const int bo = bl * KPAD + half * 2 + k0;
#pragma unroll
    for (int t = 0; t < 6; ++t) {
      v2f b = *(const v2f*)&Wt[bo + t * 16 * KPAD];
      hs[t] = WMMA_F32_16x16x4(a, b, hs[t]);
    }
  }

  // ---- out_s = silu(h_s[:, :64])  (C layout: lane->col, vgpr->row) ----
#pragma unroll
  for (int t = 0; t < 4; ++t) {
#pragma unroll
    for (int j = 0; j < 8; ++j) {
      int row = half * 8 + j;
      int n   = tileBase + row;
      if (n < N) {
        float x = hs[t][j];
        out[(long)n * DIMF + t * 16 + bl] = x * sigmoid_(x);
      }
    }
  }

  // ---- vector channels: for each c, rebuild A_v[c], 2 tiles of h_v, gate, store ----
#pragma unroll
  for (int c = 0; c < 3; ++c) {
    float ac = (c == 0) ? av0 : (c == 1) ? av1 : av2;
    __syncthreads();  // previous WMMA reads of staging done before overwrite
    for (int j = 0; j < 96; ++j) {
      int col = half * 96 + j;
      float v = 0.f;
      if (valid) {
        if (col < 64) {
          v = mrow[col] * ac * w01[col] * inv_fan;            // v01 path
        } else if (col < 96) {
          int u = col - 64;
          v = mrow[64 + u * 3 + c] * a_s * w10[u] * inv_fan;  // v10 path
        } else if (col < 160) {
          int u = col - 96;
          v = frow[u] * ac * c000;                            // sc_011 path (a_v[c] folded)
        } else {
          int u = col - 160;
          v = frow[64 + u * 3 + c] * a_s * c101;              // sc_101 path
        }
      }
      A[r * KPAD + col] = v;
    }
    __syncthreads();

    v8f hv[2] = {};
    for (int k0 = 0; k0 < KTOT; k0 += 4) {
      v2f a = *(const v2f*)&A[afo + k0];
      const int bo = bl * KPAD + half * 2 + k0 + 96 * KPAD;
#pragma unroll
      for (int t = 0; t < 2; ++t) {
        v2f b = *(const v2f*)&Wt[bo + t * 16 * KPAD];
        hv[t] = WMMA_F32_16x16x4(a, b, hv[t]);
      }
    }

    // gates live in hs[4], hs[5] at the SAME lane/vgpr slots as hv[0], hv[1]
#pragma unroll
    for (int t = 0; t < 2; ++t) {
#pragma unroll
      for (int j = 0; j < 8; ++j) {
        int row = half * 8 + j;
        int n   = tileBase + row;
        if (n < N) {
          float g = sigmoid_(hs[4 + t][j]);
          int w   = t * 16 + bl;
          out[(long)n * DIMF + NS + w * 3 + c] = g * hv[t][j];
        }
      }
    }
  }
}

extern "C" void kernel_launch(void* const* d_in, const int* in_sizes, int n_in,
                              void* d_out, int out_size, void* d_ws, size_t ws_size,
                              hipStream_t stream) {
  const float* feats = (const float*)d_in[0];
  const float* attrs = (const float*)d_in[1];
  const float* em    = (const float*)d_in[2];
  const int*   eidx  = (const int*)d_in[3];
  const float* w00   = (const float*)d_in[4];
  const float* w01   = (const float*)d_in[5];
  const float* w10   = (const float*)d_in[6];
  const float* w11   = (const float*)d_in[7];
  const float* Wls   = (const float*)d_in[8];
  const float* Wlv   = (const float*)d_in[9];
  const float* W000  = (const float*)d_in[10];
  const float* W110  = (const float*)d_in[11];
  const float* W011  = (const float*)d_in[12];
  const float* W101  = (const float*)d_in[13];
  float* out = (float*)d_out;
  float* msg = (float*)d_ws;   // N*DIMF floats of scratch

  const int N = in_sizes[0] / DIMF;
  const int E = in_sizes[3] / 2;
  const int* dst = eidx + E;   // edge_index[1]

  // 1) zero accumulator (every call: atomics accumulate)
  int zn = N * DIMF;
  zero_ws_kernel<<<(zn + 255) / 256, 256, 0, stream>>>(msg, zn);

  // 2) segment-sum via f32 global atomics (1/sqrt(5) folded in)
  int total = E * DIMF;
  edge_scatter_kernel<<<(total + 255) / 256, 256, 0, stream>>>(em, dst, msg, E);

  // 3) fused tensor-product + stacked GEMM + gating via fp32 WMMA
  int blocks = (N + NODES_PER_BLOCK - 1) / NODES_PER_BLOCK;
  size_t smem = (size_t)(NCOL + WAVES * TILE_M) * KPAD * sizeof(float);
  node_update_kernel<<<blocks, 256, smem, stream>>>(feats, attrs, msg,
                                                    w00, w01, w10, w11,
                                                    Wls, Wlv, W000, W110, W011, W101,
                                                    out, N);
}